// PolyConv_6124623364541
// MI455X (gfx1250) — compile-verified
//
#include <hip/hip_runtime.h>

#define D_FEAT 32

typedef float v2f __attribute__((ext_vector_type(2)));
typedef float v8f __attribute__((ext_vector_type(8)));

// ---------------- zero fills ----------------
__global__ void k_zero_u32(unsigned int* __restrict__ p, int n) {
    int i = blockIdx.x * blockDim.x + threadIdx.x;
    if (i < n) p[i] = 0u;
}

__global__ void k_zero_f4(float4* __restrict__ p, int n4) {
    int i = blockIdx.x * blockDim.x + threadIdx.x;
    if (i < n4) p[i] = make_float4(0.f, 0.f, 0.f, 0.f);
}

// ---------------- degree histogram (integer atomics) ----------------
__global__ void k_degree(const int* __restrict__ dst, unsigned int* __restrict__ deg,
                         int n_edges) {
    int e = blockIdx.x * blockDim.x + threadIdx.x;
    if (e < n_edges) atomicAdd(&deg[dst[e]], 1u);
}

// ---------------- d^{-1/2} ----------------
__global__ void k_dinv(const unsigned int* __restrict__ deg, float* __restrict__ dinv,
                       int n) {
    int i = blockIdx.x * blockDim.x + threadIdx.x;
    if (i < n) {
        float d = fmaxf((float)deg[i], 1.0f);
        dinv[i] = rsqrtf(d);
    }
}

// ---------------- init: f = feat, h = theta0 * feat (float4 streaming) ----------------
__global__ void k_init(const float4* __restrict__ feat, float4* __restrict__ f,
                       float4* __restrict__ h, float theta0, int n4) {
    int i = blockIdx.x * blockDim.x + threadIdx.x;
    if (i < n4) {
        float4 v = feat[i];
        f[i] = v;
        h[i] = make_float4(theta0 * v.x, theta0 * v.y, theta0 * v.z, theta0 * v.w);
    }
}

// ---------------- edge scatter: one wave32 per edge, one lane per feature ----------------
// agg[dst, lane] += f[src, lane] * dinv[src]
__global__ void k_scatter(const float* __restrict__ f, const float* __restrict__ dinv,
                          const int* __restrict__ src, const int* __restrict__ dst,
                          float* __restrict__ agg, int n_edges) {
    int gid  = blockIdx.x * blockDim.x + threadIdx.x;
    int e    = gid >> 5;
    int lane = threadIdx.x & 31;
    if (e >= n_edges) return;

    // run the index stream ahead of the gather/scatter (global_prefetch_b8)
    int pf = e + 4096;
    if (pf < n_edges) {
        __builtin_prefetch(&src[pf], 0, 0);
        __builtin_prefetch(&dst[pf], 0, 0);
    }

    int   s  = src[e];
    int   d  = dst[e];
    float di = dinv[s];                       // lane-uniform broadcast
    float v  = f[s * D_FEAT + lane] * di;     // coalesced 128B gather per wave
    unsafeAtomicAdd(&agg[d * D_FEAT + lane], v);  // global_atomic_add_f32 (no return)
}

// ---------------- node update (WMMA): one wave per 16-node x 16-feat tile ----------------
// Phase 1 (VALU, B-layout regs): f_new = f - dinv(row) * agg   (written back to f)
// Phase 2 (WMMA): H += (theta*I16) x F_new, four chained V_WMMA_F32_16X16X4_F32
//   A-matrix (theta*I slice) is synthesized in registers from the lane id — zero loads.
__global__ void k_update_wmma(float* __restrict__ f, const float* __restrict__ agg,
                              const float* __restrict__ dinv, float* __restrict__ h,
                              float theta, int n_tiles) {
    int wave = (blockIdx.x * blockDim.x + threadIdx.x) >> 5;
    int lane = threadIdx.x & 31;
    if (wave >= n_tiles) return;   // wave-uniform; active waves keep EXEC all-1s

    const int node_base = (wave >> 1) * 16;
    const int feat_base = (wave & 1) * 16;
    const int col = lane & 15;
    const int hi  = lane >> 4;     // 0: lanes 0-15, 1: lanes 16-31

    // ---- f/agg tiles in B-matrix layout (4x16 per K-chunk, 2 VGPRs each) ----
    // chunk j: VGPR0 holds row 4j+(hi?2:0), VGPR1 holds row 4j+(hi?3:1), col = lane&15
    v2f fB[4];
#pragma unroll
    for (int j = 0; j < 4; ++j) {
        int r0 = 4 * j + (hi ? 2 : 0);
        int r1 = 4 * j + (hi ? 3 : 1);
        size_t i0 = (size_t)(node_base + r0) * D_FEAT + feat_base + col;
        size_t i1 = (size_t)(node_base + r1) * D_FEAT + feat_base + col;
        float f0 = f[i0], f1 = f[i1];
        float a0 = agg[i0], a1 = agg[i1];
        float d0 = dinv[node_base + r0];   // half-wave-uniform broadcast
        float d1 = dinv[node_base + r1];
        f0 -= a0 * d0;
        f1 -= a1 * d1;
        f[i0] = f0;                        // write back f_new
        f[i1] = f1;
        fB[j].x = f0;
        fB[j].y = f1;
    }

    // ---- h tile in C/D accumulator layout: VGPR r = row r (+8 for upper half) ----
    v8f c;
    const int rowC = hi * 8;
#pragma unroll
    for (int r = 0; r < 8; ++r) {
        c[r] = h[(size_t)(node_base + r + rowC) * D_FEAT + feat_base + col];
    }

    // ---- H += theta * F_new  via  D = (theta*I) x F + C, K=16 in 4 chunks ----
    const int mA = lane & 15;   // A-matrix M index for this lane
#pragma unroll
    for (int j = 0; j < 4; ++j) {
        v2f a;
        a.x = (mA == 4 * j + (hi ? 2 : 0)) ? theta : 0.0f;  // K-slice of theta*I
        a.y = (mA == 4 * j + (hi ? 3 : 1)) ? theta : 0.0f;
        c = __builtin_amdgcn_wmma_f32_16x16x4_f32(false, a, false, fB[j],
                                                  (short)0, c, false, false);
    }

#pragma unroll
    for (int r = 0; r < 8; ++r) {
        h[(size_t)(node_base + r + rowC) * D_FEAT + feat_base + col] = c[r];
    }
}

// ---------------- scalar tail for n_nodes % 16 (float4 chunks) ----------------
__global__ void k_update_tail(float4* __restrict__ f, const float4* __restrict__ agg,
                              const float* __restrict__ dinv, float4* __restrict__ h,
                              float theta, int node_start, int n_chunks) {
    int i = blockIdx.x * blockDim.x + threadIdx.x;
    if (i >= n_chunks) return;
    int g = node_start * 8 + i;        // global float4 chunk
    int node = g >> 3;
    float di = dinv[node];
    float4 fv = f[g];
    float4 av = agg[g];
    fv.x -= av.x * di; fv.y -= av.y * di; fv.z -= av.z * di; fv.w -= av.w * di;
    f[g] = fv;
    float4 hv = h[g];
    hv.x += theta * fv.x; hv.y += theta * fv.y;
    hv.z += theta * fv.z; hv.w += theta * fv.w;
    h[g] = hv;
}

extern "C" void kernel_launch(void* const* d_in, const int* in_sizes, int n_in,
                              void* d_out, int out_size, void* d_ws, size_t ws_size,
                              hipStream_t stream) {
    const float* feat = (const float*)d_in[0];
    const int*   src  = (const int*)d_in[1];
    const int*   dst  = (const int*)d_in[2];
    float*       h    = (float*)d_out;

    const int n_nodes = in_sizes[0] / D_FEAT;
    const int n_edges = in_sizes[1];
    const int n_elems = n_nodes * D_FEAT;
    const int n4      = n_elems / 4;

    // workspace layout (256B-aligned slices)
    char*  ws  = (char*)d_ws;
    size_t off = 0;
    auto carve = [&](size_t bytes) -> void* {
        void* p = ws + off;
        off += (bytes + 255) & ~((size_t)255);
        return p;
    };
    unsigned int* deg  = (unsigned int*)carve((size_t)n_nodes * sizeof(unsigned int));
    float*        dinv = (float*)carve((size_t)n_nodes * sizeof(float));
    float*        f    = (float*)carve((size_t)n_elems * sizeof(float));
    float*        agg  = (float*)carve((size_t)n_elems * sizeof(float));
    (void)ws_size;

    const float theta[5] = {1.0f, -0.5f, 0.25f, -0.125f, 0.0625f};

    const int B = 256;
    auto blocks = [&](long long work) { return (int)((work + B - 1) / B); };

    // degree + normalization
    k_zero_u32<<<blocks(n_nodes), B, 0, stream>>>(deg, n_nodes);
    k_degree  <<<blocks(n_edges), B, 0, stream>>>(dst, deg, n_edges);
    k_dinv    <<<blocks(n_nodes), B, 0, stream>>>(deg, dinv, n_nodes);

    // h = theta0 * feat ; f = feat
    k_init<<<blocks(n4), B, 0, stream>>>((const float4*)feat, (float4*)f, (float4*)h,
                                         theta[0], n4);

    const int n16     = (n_nodes / 16) * 16;
    const int n_tiles = (n16 / 16) * 2;            // 2 feature halves per 16-node group
    const long long scatter_threads = (long long)n_edges * 32;
    const long long wmma_threads    = (long long)n_tiles * 32;

    for (int k = 1; k < 5; ++k) {
        k_zero_f4 <<<blocks(n4), B, 0, stream>>>((float4*)agg, n4);
        k_scatter <<<blocks(scatter_threads), B, 0, stream>>>(f, dinv, src, dst, agg,
                                                              n_edges);
        k_update_wmma<<<blocks(wmma_threads), B, 0, stream>>>(f, agg, dinv, h,
                                                              theta[k], n_tiles);
        if (n_nodes > n16) {
            int tail_chunks = (n_nodes - n16) * 8;
            k_update_tail<<<blocks(tail_chunks), B, 0, stream>>>(
                (float4*)f, (const float4*)agg, dinv, (float4*)h, theta[k],
                n16, tail_chunks);
        }
    }
}